// ScaledDotProductAttention_16381005267233
// MI455X (gfx1250) — compile-verified
//
#include <hip/hip_runtime.h>

typedef __attribute__((ext_vector_type(2))) float v2f;
typedef __attribute__((ext_vector_type(4))) float v4f;
typedef __attribute__((ext_vector_type(8))) float v8f;

#define Bb 4
#define Hh 16
#define Ss 2048
#define Dd 64
#define ROWS 32
#define LDS_STRIDE 2052  // 2048 + 4 pad -> conflict-free 16x4 f32 A-fragment reads

__device__ __forceinline__ float wave_reduce_max(float v) {
#pragma unroll
  for (int off = 16; off > 0; off >>= 1) v = fmaxf(v, __shfl_xor(v, off, 32));
  return v;
}
__device__ __forceinline__ float wave_reduce_sum(float v) {
#pragma unroll
  for (int off = 16; off > 0; off >>= 1) v += __shfl_xor(v, off, 32);
  return v;
}

__global__ __launch_bounds__(256) void sdpa_fp32_wmma_kernel(
    const float* __restrict__ Q, const float* __restrict__ K,
    const float* __restrict__ V, const float* __restrict__ mask,
    float* __restrict__ ctx, float* __restrict__ probs) {
  extern __shared__ float sm[];  // [ROWS][LDS_STRIDE] score/prob tile

  const int b = blockIdx.z, h = blockIdx.y, qb = blockIdx.x;
  const int qbase = qb * ROWS;
  const int tid  = threadIdx.x;
  const int lane = tid & 31;
  const int w    = tid >> 5;     // wave id 0..7
  const int m    = lane & 15;    // A-row / B-col index within fragment
  const int half = lane >> 4;    // lane half selects k-pair

  const size_t bh = (size_t)b * Hh + h;
  const float* Qbh = Q + bh * (size_t)Ss * Dd;
  const float* Kbh = K + bh * (size_t)Ss * Dd;
  const float* Vbh = V + bh * (size_t)Ss * Dd;

  const int rt = w >> 2;  // which 16-row tile of the 32-row block (0/1)

  // ---- preload Q A-fragments: 16 chained k-steps cover K=64
  v2f qf[16];
  {
    const float* qrow = Qbh + (size_t)(qbase + rt * 16 + m) * Dd + 2 * half;
#pragma unroll
    for (int ks = 0; ks < 16; ++ks) qf[ks] = *(const v2f*)(qrow + 4 * ks);
  }

  // ---- phase 1: scores = (1/sqrt(64)) * Q K^T  -> LDS
  const float scale = 0.125f;
  for (int ct = (w & 3); ct < Ss / 16; ct += 4) {
    const float* krow = Kbh + (size_t)(ct * 16 + m) * Dd + 2 * half;
    v8f c = {};
#pragma unroll
    for (int ks = 0; ks < 16; ++ks) {
      v2f kf = *(const v2f*)(krow + 4 * ks);
      c = __builtin_amdgcn_wmma_f32_16x16x4_f32(false, qf[ks], false, kf,
                                                (short)0, c, false, false);
    }
#pragma unroll
    for (int vv = 0; vv < 8; ++vv)
      sm[(rt * 16 + vv + 8 * half) * LDS_STRIDE + ct * 16 + m] = c[vv] * scale;
  }
  __syncthreads();

  // ---- phase 2: exact softmax per row (+ additive mask), stream probs out
  const v4f* mask4 = (const v4f*)(mask + (size_t)b * Ss);
  for (int r = w * 4; r < w * 4 + 4; ++r) {
    v4f* row4 = (v4f*)(sm + (size_t)r * LDS_STRIDE);
    float mx = -3.4e38f;
    for (int i = lane; i < Ss / 4; i += 32) {
      v4f s = row4[i] + mask4[i];
      row4[i] = s;
      mx = fmaxf(fmaxf(fmaxf(mx, s[0]), fmaxf(s[1], s[2])), s[3]);
    }
    mx = wave_reduce_max(mx);
    float l = 0.f;
    for (int i = lane; i < Ss / 4; i += 32) {
      v4f s = row4[i];
      v4f e;
      e[0] = __expf(s[0] - mx); e[1] = __expf(s[1] - mx);
      e[2] = __expf(s[2] - mx); e[3] = __expf(s[3] - mx);
      row4[i] = e;
      l += e[0] + e[1] + e[2] + e[3];
    }
    l = wave_reduce_sum(l);
    const float inv = 1.0f / l;
    v4f* prow4 = (v4f*)(probs + (bh * Ss + (size_t)(qbase + r)) * Ss);
    for (int i = lane; i < Ss / 4; i += 32) {
      v4f p = row4[i] * inv;
      row4[i]  = p;   // keep normalized P in LDS for the P*V GEMM
      prow4[i] = p;   // bandwidth-critical global store
    }
  }
  __syncthreads();

  // ---- phase 3: context = P (LDS) * V (global); 8 waves = 2x4 tiles of 16x16
  const int ctile = w & 3;
  const float* vcol = Vbh + ctile * 16 + m;                      // B-fragment: col m of V
  const float* prow = sm + (size_t)(rt * 16 + m) * LDS_STRIDE + 2 * half;  // A-fragment
  v8f acc = {};
#pragma unroll 4
  for (int kb = 0; kb < Ss; kb += 4) {
    v2f af = *(const v2f*)(prow + kb);
    v2f bf;
    bf[0] = vcol[(size_t)(kb + 2 * half) * Dd];
    bf[1] = vcol[(size_t)(kb + 2 * half + 1) * Dd];
    acc = __builtin_amdgcn_wmma_f32_16x16x4_f32(false, af, false, bf,
                                                (short)0, acc, false, false);
  }
#pragma unroll
  for (int vv = 0; vv < 8; ++vv) {
    const int q = qbase + rt * 16 + vv + 8 * half;
    ctx[(bh * Ss + (size_t)q) * Dd + ctile * 16 + m] = acc[vv];
  }
}

extern "C" void kernel_launch(void* const* d_in, const int* in_sizes, int n_in,
                              void* d_out, int out_size, void* d_ws, size_t ws_size,
                              hipStream_t stream) {
  const float* Q    = (const float*)d_in[0];
  const float* K    = (const float*)d_in[1];
  const float* V    = (const float*)d_in[2];
  const float* mask = (const float*)d_in[3];

  float* ctx   = (float*)d_out;                                // context first
  float* probs = ctx + (size_t)Bb * Hh * Ss * Dd;              // then probs

  dim3 grid(Ss / ROWS, Hh, Bb);
  dim3 block(256);
  size_t lds_bytes = (size_t)ROWS * LDS_STRIDE * sizeof(float);  // 262,656 B < 320 KB
  sdpa_fp32_wmma_kernel<<<grid, block, lds_bytes, stream>>>(Q, K, V, mask, ctx, probs);
}